// GraphActorCriticEQL_74775380624041
// MI455X (gfx1250) — compile-verified
//
#include <hip/hip_runtime.h>

typedef __attribute__((ext_vector_type(2))) float v2f;
typedef __attribute__((ext_vector_type(8))) float v8f;

#define NB 256
#define NN 128
#define NA 15
#define NH 32
#define MID 256

// ---------------------------------------------------------------------------
// Phase 1: messenger. msg[b,i] = sum_j relu(u[i]+v[j]) . mw2  + N*mb2
// ---------------------------------------------------------------------------
__global__ __launch_bounds__(256)
void eql_msg_kernel(const float* __restrict__ obs,
                    const float* __restrict__ mw1, const float* __restrict__ mb1,
                    const float* __restrict__ mw2, const float* __restrict__ mb2,
                    float* __restrict__ msg_out) {
  __shared__ float Us[NN][NH];
  __shared__ float Vs[NN][NH];
  __shared__ float W2[NH];
  __shared__ float partial[256];
  const int b = blockIdx.x;
  const int t = threadIdx.x;
  if (t < NH) W2[t] = mw2[t];
  for (int e = t; e < NN * NH; e += 256) {
    const int i = e >> 5, h = e & 31;
    const float o = obs[b * NN + i];
    const float c = (float)i * (1.0f / (float)NN);
    Us[i][h] = o * mw1[0 * NH + h] + c * mw1[1 * NH + h] + mb1[h];
    Vs[i][h] = o * mw1[2 * NH + h] + c * mw1[3 * NH + h];
  }
  __syncthreads();
  const int i  = t >> 1;         // node handled by this thread (2 threads/node)
  const int j0 = (t & 1) * 64;   // half of the j range
  float ui[NH];
#pragma unroll
  for (int h = 0; h < NH; ++h) ui[h] = Us[i][h];
  float acc = 0.0f;
  for (int j = j0; j < j0 + 64; ++j) {
#pragma unroll
    for (int h = 0; h < NH; ++h) {
      const float p = ui[h] + Vs[j][h];
      acc = fmaf(fmaxf(p, 0.0f), W2[h], acc);
    }
  }
  partial[t] = acc;
  __syncthreads();
  if ((t & 1) == 0)
    msg_out[b * NN + i] = partial[t] + partial[t + 1] + (float)NN * mb2[0];
}

// ---------------------------------------------------------------------------
// Phase 2: actor. logits[b,a] = sum_i relu(msg_i*w0 + coor_i*w1 + a*w2a + ab1) . aw2 + N*ab2
// ---------------------------------------------------------------------------
__global__ __launch_bounds__(256)
void eql_actor_kernel(const float* __restrict__ msg,
                      const float* __restrict__ aw1, const float* __restrict__ ab1,
                      const float* __restrict__ aw2, const float* __restrict__ ab2,
                      float* __restrict__ logits) {
  __shared__ float PA[NA][NH];
  __shared__ float W0[NH], W1[NH], W2[NH];
  __shared__ float M[NN];
  __shared__ float SL[NA];
  const int b = blockIdx.x, t = threadIdx.x;
  if (t < NH) { W0[t] = aw1[0 * NH + t]; W1[t] = aw1[1 * NH + t]; W2[t] = aw2[t]; }
  if (t < NA) SL[t] = 0.0f;
  for (int e = t; e < NA * NH; e += 256) {
    const int a = e >> 5, h = e & 31;
    PA[a][h] = (float)a * aw1[2 * NH + h] + ab1[h];
  }
  for (int e = t; e < NN; e += 256) M[e] = msg[b * NN + e];
  __syncthreads();
  for (int p = t; p < NN * NA; p += 256) {
    const int i = p / NA, a = p % NA;
    const float mi = M[i], ci = (float)i * (1.0f / (float)NN);
    float acc = 0.0f;
#pragma unroll
    for (int h = 0; h < NH; ++h) {
      const float pre = fmaf(mi, W0[h], fmaf(ci, W1[h], PA[a][h]));
      acc = fmaf(fmaxf(pre, 0.0f), W2[h], acc);
    }
    atomicAdd(&SL[a], acc);
  }
  __syncthreads();
  if (t < NA) logits[b * NA + t] = SL[t] + (float)NN * ab2[0];
}

// ---------------------------------------------------------------------------
// Phase 3: critic MLP via V_WMMA_F32_16X16X4_F32 (exact f32 matrix path).
// 16 batch rows per block, 16 waves; wave w owns output columns [16w,16w+16).
// ---------------------------------------------------------------------------
template <int K, int INS, int OUTS>
__device__ inline void wmma_layer_f32(const float (*__restrict__ Xin)[INS],
                                      float (*__restrict__ Xout)[OUTS],
                                      const float* __restrict__ W,
                                      const float* __restrict__ bias,
                                      int nl, int half, int ncol) {
  v8f c = {};
  for (int k = 0; k < K; k += 4) {
    const int ka = k + 2 * half;                 // ISA A layout: K = 2*half + {0,1}
    v2f A;  A.x  = Xin[nl][ka];       A.y  = Xin[nl][ka + 1];
    v2f Bv; Bv.x = W[ka * MID + ncol]; Bv.y = W[(ka + 1) * MID + ncol];
    c = __builtin_amdgcn_wmma_f32_16x16x4_f32(false, A, false, Bv,
                                              (short)0, c, false, false);
  }
  const float bs = bias[ncol];
#pragma unroll
  for (int r = 0; r < 8; ++r)                    // D: vgpr r -> M = r + 8*half
    Xout[r + 8 * half][ncol] = fmaxf(c[r] + bs, 0.0f);
}

__global__ __launch_bounds__(512)
void critic_wmma_kernel(const float* __restrict__ obs,
                        const float* __restrict__ cw1, const float* __restrict__ cb1,
                        const float* __restrict__ cw2, const float* __restrict__ cb2,
                        const float* __restrict__ cw3, const float* __restrict__ cb3,
                        const float* __restrict__ cw4, const float* __restrict__ cb4,
                        float* __restrict__ value) {
  constexpr int XS = NN + 4;    // stride pad: (stride mod 64 banks) = 4 -> conflict-free
  constexpr int HS = MID + 4;
  __shared__ float Xs[16][XS];
  __shared__ float Ha[16][HS];
  __shared__ float Hb[16][HS];
  __shared__ float sVal[16];
  const int t    = threadIdx.x;
  const int wave = t >> 5;
  const int lane = t & 31;
  const int half = lane >> 4;
  const int nl   = lane & 15;
  const int ncol = wave * 16 + nl;               // this lane's output column
  const int row0 = blockIdx.x * 16;

  for (int e = t; e < 16 * NN; e += 512) {
    const int m = e >> 7, k = e & 127;
    Xs[m][k] = obs[(row0 + m) * NN + k];
  }
  if (t < 16) sVal[t] = 0.0f;
  __syncthreads();

  wmma_layer_f32<NN,  XS, HS>(Xs, Ha, cw1, cb1, nl, half, ncol);   // 128 -> 256
  __syncthreads();
  wmma_layer_f32<MID, HS, HS>(Ha, Hb, cw2, cb2, nl, half, ncol);   // 256 -> 256
  __syncthreads();
  wmma_layer_f32<MID, HS, HS>(Hb, Ha, cw3, cb3, nl, half, ncol);   // 256 -> 256
  __syncthreads();

  // final 256 -> 1: wave w reduces row w
  {
    const int m = wave;
    float p = 0.0f;
    for (int k = lane; k < MID; k += 32) p = fmaf(Ha[m][k], cw4[k], p);
    atomicAdd(&sVal[m], p);
  }
  __syncthreads();
  if (t < 16) value[row0 + t] = sVal[t] + cb4[0];
}

// ---------------------------------------------------------------------------
extern "C" void kernel_launch(void* const* d_in, const int* in_sizes, int n_in,
                              void* d_out, int out_size, void* d_ws, size_t ws_size,
                              hipStream_t stream) {
  const float* obs = (const float*)d_in[0];
  const float* mw1 = (const float*)d_in[1];
  const float* mb1 = (const float*)d_in[2];
  const float* mw2 = (const float*)d_in[3];
  const float* mb2 = (const float*)d_in[4];
  const float* aw1 = (const float*)d_in[5];
  const float* ab1 = (const float*)d_in[6];
  const float* aw2 = (const float*)d_in[7];
  const float* ab2 = (const float*)d_in[8];
  const float* cw1 = (const float*)d_in[9];
  const float* cb1 = (const float*)d_in[10];
  const float* cw2 = (const float*)d_in[11];
  const float* cb2 = (const float*)d_in[12];
  const float* cw3 = (const float*)d_in[13];
  const float* cb3 = (const float*)d_in[14];
  const float* cw4 = (const float*)d_in[15];
  const float* cb4 = (const float*)d_in[16];

  float* logits = (float*)d_out;            // [256,15]
  float* value  = (float*)d_out + NB * NA;  // [256]
  float* msg    = (float*)d_ws;             // [256,128] scratch

  eql_msg_kernel<<<NB, 256, 0, stream>>>(obs, mw1, mb1, mw2, mb2, msg);
  eql_actor_kernel<<<NB, 256, 0, stream>>>(msg, aw1, ab1, aw2, ab2, logits);
  critic_wmma_kernel<<<NB / 16, 512, 0, stream>>>(obs, cw1, cb1, cw2, cb2,
                                                  cw3, cb3, cw4, cb4, value);
}